// DetNMSPostProcessor_58317065945401
// MI455X (gfx1250) — compile-verified
//
#include <hip/hip_runtime.h>
#include <hip/hip_bf16.h>
#include <stdint.h>

#define NQ   1000
#define NP   1024
#define NC   80
#define TOPK_OUT 300
#define NT   256
#define IOU_T   0.01f
#define SCORE_T 0.1f
#define IMG_SCALE 640.0f
#define LOG2E 1.44269504088896340736f

typedef unsigned int u32x4 __attribute__((ext_vector_type(4)));
typedef int          i32x4 __attribute__((ext_vector_type(4)));
typedef int          i32x8 __attribute__((ext_vector_type(8)));

__device__ __forceinline__ float fast_rcp(float x) { return __builtin_amdgcn_rcpf(x); }
__device__ __forceinline__ float fast_sigmoid(float x) {
  return fast_rcp(1.0f + __builtin_amdgcn_exp2f(-x * LOG2E));
}
__device__ __forceinline__ unsigned score_key_hi(float s) {
  unsigned u = __float_as_uint(s);
  u = (u & 0x80000000u) ? ~u : (u | 0x80000000u);
  return ~u;
}

// Scratch layout: 9 arrays of B*NP elements (4B each):
// 0 scores(f) 1 labels(i) 2 bx0 3 by0 4 bx1 5 by1 6 order(i) 7 keep(i) 8 maxc(f)
#define WS_SCORES 0
#define WS_LABELS 1
#define WS_BX0    2
#define WS_BY0    3
#define WS_BX1    4
#define WS_BY1    5
#define WS_ORDER  6
#define WS_KEEP   7
#define WS_MAXC   8
#define WS_SLOTS  9

// ---------------------------------------------------------------------------
// kA: class argmax + sigmoid scores
// ---------------------------------------------------------------------------
__global__ __launch_bounds__(NT)
void kA_scores(const float* __restrict__ logits, float* __restrict__ ws, int B)
{
  const int b = blockIdx.x, tid = threadIdx.x;
  const float* lg = logits + (size_t)b * NQ * NC;
  float* scores = ws + (size_t)WS_SCORES * B * NP + (size_t)b * NP;
  int*   labels = (int*)ws + (size_t)WS_LABELS * B * NP + (size_t)b * NP;
  for (int p = 0; p < 4; ++p) {
    int q = tid + p * NT;
    if (q < NQ) {
      const float* row = lg + (size_t)q * NC;
      float best = -3.0e38f; int bi = 0;
      for (int c = 0; c < NC; ++c) {
        float v = row[c];
        if (v > best) { best = v; bi = c; }
      }
      scores[q] = fast_sigmoid(best);
      labels[q] = bi;
    } else {
      scores[q] = 0.0f;
      labels[q] = 0;
    }
  }
}

// ---------------------------------------------------------------------------
// kB: TDM-stage raw cxcywh boxes into LDS (TENSOR_LOAD_TO_LDS, TENSORcnt),
//     then transform to xyxy*640 + per-image max coordinate (LDS reduce).
// ---------------------------------------------------------------------------
__global__ __launch_bounds__(NT)
void kB_boxes(const float* __restrict__ boxes, float* __restrict__ ws, int B)
{
  __shared__ float s_stage[NQ * 4];   // 16000 B staged by the Tensor Data Mover
  __shared__ float s_red[NT];
  const int b = blockIdx.x, tid = threadIdx.x;
  const float* bx = boxes + (size_t)b * NQ * 4;
  float* bx0 = ws + (size_t)WS_BX0 * B * NP + (size_t)b * NP;
  float* by0 = ws + (size_t)WS_BY0 * B * NP + (size_t)b * NP;
  float* bx1 = ws + (size_t)WS_BX1 * B * NP + (size_t)b * NP;
  float* by1 = ws + (size_t)WS_BY1 * B * NP + (size_t)b * NP;
  float* maxc = ws + (size_t)WS_MAXC * B * NP;

#if __has_builtin(__builtin_amdgcn_tensor_load_to_lds)
  if (tid < 32) {  // TDM issues per wave (EXEC ignored): wave 0 only
    unsigned ldsAddr = (unsigned)(uintptr_t)(&s_stage[0]);
    unsigned long long ga = (unsigned long long)(uintptr_t)bx;

    // D# group 0: count=1 | lds_addr | global_addr[56:0] | type=2
    u32x4 g0;
    g0[0] = 1u;
    g0[1] = ldsAddr;
    g0[2] = (unsigned)(ga & 0xFFFFFFFFull);
    g0[3] = (unsigned)((ga >> 32) & 0x01FFFFFFull) | (2u << 30);

    // D# group 1: data_size=4B; 1-D tile: tile_dim0 = tensor_dim0 = 4000 elems
    const unsigned td0 = 4000u, td1 = 1u, tile0 = 4000u, stride0 = 4000u;
    i32x8 g1;
    g1[0] = (int)(2u << 16);                               // data_size=2 (4 B)
    g1[1] = (int)((td0 & 0xFFFFu) << 16);                  // tensor_dim0 lo16
    g1[2] = (int)((td0 >> 16) | ((td1 & 0xFFFFu) << 16));  // td0 hi | td1 lo
    g1[3] = (int)((td1 >> 16) | (tile0 << 16));            // td1 hi | tile_dim0
    g1[4] = 0;                                             // tile_dim1/2 unused
    g1[5] = (int)stride0;                                  // dim0 stride lo32
    g1[6] = 0;
    g1[7] = 0;

    i32x4 gz4 = {0, 0, 0, 0};
    i32x8 gz8 = {0, 0, 0, 0, 0, 0, 0, 0};
    __builtin_amdgcn_tensor_load_to_lds(g0, g1, gz4, gz4, gz8, 0);
    __builtin_amdgcn_s_wait_tensorcnt(0);
  }
#else
  for (int p = 0; p < 4; ++p) {
    int q = tid + p * NT;
    if (q < NQ) {
      s_stage[q * 4 + 0] = bx[q * 4 + 0];
      s_stage[q * 4 + 1] = bx[q * 4 + 1];
      s_stage[q * 4 + 2] = bx[q * 4 + 2];
      s_stage[q * 4 + 3] = bx[q * 4 + 3];
    }
  }
#endif
  __syncthreads();

  float localMax = -3.0e38f;
  for (int p = 0; p < 4; ++p) {
    int q = tid + p * NT;
    if (q < NQ) {
      float cx = s_stage[q * 4 + 0];
      float cy = s_stage[q * 4 + 1];
      float w  = s_stage[q * 4 + 2];
      float h  = s_stage[q * 4 + 3];
      float x0 = (cx - 0.5f * w) * IMG_SCALE;
      float y0 = (cy - 0.5f * h) * IMG_SCALE;
      float x1 = (cx + 0.5f * w) * IMG_SCALE;
      float y1 = (cy + 0.5f * h) * IMG_SCALE;
      bx0[q] = x0; by0[q] = y0; bx1[q] = x1; by1[q] = y1;
      localMax = fmaxf(localMax, fmaxf(fmaxf(x0, y0), fmaxf(x1, y1)));
    } else {
      bx0[q] = 0.0f; by0[q] = 0.0f; bx1[q] = 0.0f; by1[q] = 0.0f;
    }
  }
  s_red[tid] = localMax;
  __syncthreads();
  for (int s = NT / 2; s > 0; s >>= 1) {
    if (tid < s) s_red[tid] = fmaxf(s_red[tid], s_red[tid + s]);
    __syncthreads();
  }
  if (tid == 0) maxc[b] = s_red[0];
}

// ---------------------------------------------------------------------------
// kC: build keys + bitonic sort (LDS) -> sorted original indices ("order")
// ---------------------------------------------------------------------------
__global__ __launch_bounds__(NT)
void kC_sort(float* __restrict__ ws, int B)
{
  __shared__ unsigned s_khi[NP], s_klo[NP];
  const int b = blockIdx.x, tid = threadIdx.x;
  const float* scores = ws + (size_t)WS_SCORES * B * NP + (size_t)b * NP;
  int* order = (int*)ws + (size_t)WS_ORDER * B * NP + (size_t)b * NP;

  for (int p = 0; p < 4; ++p) {
    int q = tid + p * NT;
    unsigned hi;
    if (q < NQ) {
      float sc = scores[q];
      float s  = (sc > SCORE_T) ? sc : -1.0f;
      hi = score_key_hi(s);
    } else {
      hi = 0xFFFFFFFFu;
    }
    s_khi[q] = hi;
    s_klo[q] = (unsigned)q;
  }
  __syncthreads();

  for (int k = 2; k <= NP; k <<= 1) {
    for (int j = k >> 1; j > 0; j >>= 1) {
      for (int p = 0; p < 4; ++p) {
        int i = tid + p * NT;
        int ixj = i ^ j;
        if (ixj > i) {
          unsigned ahi = s_khi[i],   alo = s_klo[i];
          unsigned chi = s_khi[ixj], clo = s_klo[ixj];
          bool agtc = (ahi > chi) || ((ahi == chi) && (alo > clo));
          bool up = ((i & k) == 0);
          if (agtc == up) {
            s_khi[i] = chi; s_klo[i] = clo;
            s_khi[ixj] = ahi; s_klo[ixj] = alo;
          }
        }
      }
      __syncthreads();
    }
  }

  for (int p = 0; p < 4; ++p) {
    int q = tid + p * NT;
    order[q] = (int)s_klo[q];
  }
}

// ---------------------------------------------------------------------------
// kD: greedy NMS over class-offset boxes in sorted order
// ---------------------------------------------------------------------------
__global__ __launch_bounds__(NT)
void kD_nms(float* __restrict__ ws, int B)
{
  __shared__ float s_ox0[NP], s_oy0[NP], s_ox1[NP], s_oy1[NP];
  __shared__ int   s_keepi[NP];
  const int b = blockIdx.x, tid = threadIdx.x;
  const float* scores = ws + (size_t)WS_SCORES * B * NP + (size_t)b * NP;
  const int*   labels = (const int*)ws + (size_t)WS_LABELS * B * NP + (size_t)b * NP;
  const float* bx0 = ws + (size_t)WS_BX0 * B * NP + (size_t)b * NP;
  const float* by0 = ws + (size_t)WS_BY0 * B * NP + (size_t)b * NP;
  const float* bx1 = ws + (size_t)WS_BX1 * B * NP + (size_t)b * NP;
  const float* by1 = ws + (size_t)WS_BY1 * B * NP + (size_t)b * NP;
  const int* order = (const int*)ws + (size_t)WS_ORDER * B * NP + (size_t)b * NP;
  int* keep = (int*)ws + (size_t)WS_KEEP * B * NP + (size_t)b * NP;
  const float maxc = (ws + (size_t)WS_MAXC * B * NP)[b];

  for (int p = 0; p < 4; ++p) {
    int jj = tid + p * NT;
    if (jj < NQ) {
      int o = order[jj];
      float off = (float)labels[o] * (maxc + 1.0f);
      s_ox0[jj] = bx0[o] + off;
      s_oy0[jj] = by0[o] + off;
      s_ox1[jj] = bx1[o] + off;
      s_oy1[jj] = by1[o] + off;
      s_keepi[jj] = (scores[o] > SCORE_T) ? 1 : 0;
    } else {
      s_ox0[jj] = 0.0f; s_oy0[jj] = 0.0f; s_ox1[jj] = 0.0f; s_oy1[jj] = 0.0f;
      s_keepi[jj] = 0;
    }
  }

  for (int i = 0; i < NQ; ++i) {
    __syncthreads();
    if (s_keepi[i] != 0) {
      float ax0 = s_ox0[i], ay0 = s_oy0[i], ax1 = s_ox1[i], ay1 = s_oy1[i];
      float areai = fmaxf(ax1 - ax0, 0.0f) * fmaxf(ay1 - ay0, 0.0f);
      for (int jj = i + 1 + tid; jj < NQ; jj += NT) {
        if (s_keepi[jj] != 0) {
          float b0 = s_ox0[jj], b1 = s_oy0[jj], b2 = s_ox1[jj], b3 = s_oy1[jj];
          float areaj = fmaxf(b2 - b0, 0.0f) * fmaxf(b3 - b1, 0.0f);
          float iw = fmaxf(fminf(ax1, b2) - fmaxf(ax0, b0), 0.0f);
          float ih = fmaxf(fminf(ay1, b3) - fmaxf(ay0, b1), 0.0f);
          float inter = iw * ih;
          float uni = areai + areaj - inter;
          float iou = inter * fast_rcp(fmaxf(uni, 1e-9f));
          if (iou > IOU_T) s_keepi[jj] = 0;
        }
      }
    }
  }
  __syncthreads();

  for (int p = 0; p < 4; ++p) {
    int jj = tid + p * NT;
    keep[jj] = s_keepi[jj];
  }
}

// ---------------------------------------------------------------------------
// kE: compact first TOPK kept + write outputs
// ---------------------------------------------------------------------------
__global__ __launch_bounds__(NT)
void kE_output(const float* __restrict__ ws_c, float* __restrict__ out, int B)
{
  __shared__ int s_take[TOPK_OUT];
  __shared__ int s_nkept;
  const int b = blockIdx.x, tid = threadIdx.x;
  const float* ws = ws_c;
  const float* scores = ws + (size_t)WS_SCORES * B * NP + (size_t)b * NP;
  const int*   labels = (const int*)ws + (size_t)WS_LABELS * B * NP + (size_t)b * NP;
  const float* bx0 = ws + (size_t)WS_BX0 * B * NP + (size_t)b * NP;
  const float* by0 = ws + (size_t)WS_BY0 * B * NP + (size_t)b * NP;
  const float* bx1 = ws + (size_t)WS_BX1 * B * NP + (size_t)b * NP;
  const float* by1 = ws + (size_t)WS_BY1 * B * NP + (size_t)b * NP;
  const int* order = (const int*)ws + (size_t)WS_ORDER * B * NP + (size_t)b * NP;
  const int* keep  = (const int*)ws + (size_t)WS_KEEP * B * NP + (size_t)b * NP;

  if (tid == 0) {
    int cnt = 0;
    for (int jj = 0; jj < NQ && cnt < TOPK_OUT; ++jj) {
      if (keep[jj] != 0) s_take[cnt++] = jj;
    }
    s_nkept = cnt;
  }
  __syncthreads();
  const int nk = s_nkept;

  float* outBoxes  = out + (size_t)b * TOPK_OUT * 4;
  float* outScores = out + (size_t)B * TOPK_OUT * 4 + (size_t)b * TOPK_OUT;
  float* outLabels = out + (size_t)B * TOPK_OUT * 5 + (size_t)b * TOPK_OUT;
  float* outValid  = out + (size_t)B * TOPK_OUT * 6 + (size_t)b * TOPK_OUT;
  for (int k = tid; k < TOPK_OUT; k += NT) {
    float ox0 = 0.0f, oy0 = 0.0f, ox1 = 0.0f, oy1 = 0.0f;
    float sc = 0.0f, lb = -1.0f, vv = 0.0f;
    if (k < nk) {
      int jj = s_take[k];
      int o  = order[jj];
      ox0 = bx0[o]; oy0 = by0[o]; ox1 = bx1[o]; oy1 = by1[o];
      sc = scores[o];
      lb = (float)labels[o];
      vv = 1.0f;
    }
    outBoxes[(size_t)k * 4 + 0] = ox0;
    outBoxes[(size_t)k * 4 + 1] = oy0;
    outBoxes[(size_t)k * 4 + 2] = ox1;
    outBoxes[(size_t)k * 4 + 3] = oy1;
    outScores[k] = sc;
    outLabels[k] = lb;
    outValid[k]  = vv;
  }
}

extern "C" void kernel_launch(void* const* d_in, const int* in_sizes, int n_in,
                              void* d_out, int out_size, void* d_ws, size_t ws_size,
                              hipStream_t stream) {
  const float* logits = (const float*)d_in[0];  // [B,1000,80]
  const float* boxes  = (const float*)d_in[1];  // [B,1000,4]
  float* out = (float*)d_out;
  float* ws  = (float*)d_ws;
  int B = in_sizes[0] / (NQ * NC);
  size_t need = (size_t)WS_SLOTS * B * NP * sizeof(float);

  if (ws_size >= need) {
    kA_scores<<<dim3(B), dim3(NT), 0, stream>>>(logits, ws, B);
    kB_boxes <<<dim3(B), dim3(NT), 0, stream>>>(boxes, ws, B);
    kC_sort  <<<dim3(B), dim3(NT), 0, stream>>>(ws, B);
    kD_nms   <<<dim3(B), dim3(NT), 0, stream>>>(ws, B);
    kE_output<<<dim3(B), dim3(NT), 0, stream>>>(ws, out, B);
  }
}